// Hausdorff_30416958390582
// MI455X (gfx1250) — compile-verified
//
#include <hip/hip_runtime.h>
#include <hip/hip_bf16.h>

typedef __attribute__((ext_vector_type(2))) float v2f;
typedef __attribute__((ext_vector_type(8))) float v8f;

#define V 8000
#define NMASKW 250      // 8000 / 32 bit-words
#define MASK_PAD 256    // padded words per mask
#define SENTINEL 1e18f  // sqrt(1e18) ~= 1e9 == reference BIG
#define ROWTILES 500    // 8000 / 16
#define PART_STRIDE 512
#define WAVES_PB 4      // row tiles per block
#define BLOCK2 128

__device__ __forceinline__ float coordc(int i, int which) {
    // coords[v] = (v/400, (v/20)%20, v%20) / 20
    int c = (which == 0) ? (i / 400) : (which == 1) ? ((i / 20) % 20) : (i % 20);
    return (float)c * (1.0f / 20.0f);
}

// ---------------- kernel 0: zero the counters ----------------
__global__ __launch_bounds__(32) void init_kernel(int* cnt) {
    if (threadIdx.x < 8) cnt[threadIdx.x] = 0;
}

// ---------------- kernel 1: binarize -> bitmasks + counts ----------------
__global__ __launch_bounds__(256) void mask_kernel(const float* __restrict__ predict,
                                                   const float* __restrict__ target,
                                                   unsigned* __restrict__ maskA,
                                                   unsigned* __restrict__ maskB,
                                                   int* __restrict__ cnt) {
    int t = blockIdx.x * 256 + threadIdx.x;
    if (t >= 2 * V) return;                 // V % 32 == 0: whole waves exit uniformly
    int s = t / V;
    int i = t - s * V;
    bool a = rintf(predict[t]) != 0.0f;     // round-half-even like jnp.round
    bool b = rintf(target[t])  != 0.0f;
    unsigned ba = __builtin_amdgcn_ballot_w32(a);
    unsigned bb = __builtin_amdgcn_ballot_w32(b);
    if ((threadIdx.x & 31) == 0) {
        int w = i >> 5;
        maskA[s * MASK_PAD + w] = ba;
        maskB[s * MASK_PAD + w] = bb;
        atomicAdd(&cnt[s * 4 + 0], __popc(ba));         // |A|
        atomicAdd(&cnt[s * 4 + 1], __popc(ba & ~bb));   // |A \ B|
        atomicAdd(&cnt[s * 4 + 2], __popc(bb & ~ba));   // |B \ A|
    }
}

// ---------------- kernel 2: WMMA-tiled directed-distance pass ----------------
// grid = (125 blocks of 4 waves -> 500 row tiles, 4 sample/direction combos).
// D = A x B + C with A=(-2x,-2y,-2z,1), B=(xc,yc,zc,|c|^2+bias), C=|row|^2
//   => D = d^2 + (inSet ? 0 : 1e18); min-reduce, no post-masking needed.
__global__ __launch_bounds__(BLOCK2) void haus_min_kernel(const unsigned* __restrict__ maskA,
                                                          const unsigned* __restrict__ maskB,
                                                          float* __restrict__ partial) {
    __shared__ unsigned colMask[MASK_PAD];  // 1 KB
    __shared__ float4 tab[400];             // 6.4 KB: {y, z, y^2+z^2, 0} for r = c % 400

    const int sdi = blockIdx.y;
    const int s = sdi >> 1;
    const int dir = sdi & 1;
    const unsigned* mA = maskA + s * MASK_PAD;
    const unsigned* mB = maskB + s * MASK_PAD;
    const unsigned* cm = (dir == 0) ? mB : mA;   // column set we min over

    const int tid = threadIdx.x;
    for (int w = tid; w < NMASKW; w += BLOCK2) colMask[w] = cm[w];
    for (int r = tid; r < 400; r += BLOCK2) {
        float y = (float)(r / 20) * (1.0f / 20.0f);
        float z = (float)(r % 20) * (1.0f / 20.0f);
        tab[r] = make_float4(y, z, y * y + z * z, 0.0f);
    }
    __syncthreads();

    const int lane = tid & 31;
    const int half = lane >> 4;       // 0: K=0,1 slice   1: K=2,3 slice
    const int nSlot = lane & 15;
    const int rowTile = blockIdx.x * WAVES_PB + (tid >> 5);
    const int rowBase = rowTile * 16;

    // A operand (16x4 f32): lane holds row M = lane&15; VGPR0/1 = K{0,1} | K{2,3}
    const int arow = rowBase + nSlot;
    const float ax = coordc(arow, 0), ay = coordc(arow, 1), az = coordc(arow, 2);
    v2f A;
    A.x = half ? (-2.0f * az) : (-2.0f * ax);
    A.y = half ? 1.0f : (-2.0f * ay);

    // Loop-invariant C: VGPR r holds |row M|^2 with M = r + 8*half
    v8f C;
#pragma unroll
    for (int r = 0; r < 8; ++r) {
        int rr = rowBase + r + 8 * half;
        float x = coordc(rr, 0), y = coordc(rr, 1), z = coordc(rr, 2);
        C[r] = x * x + y * y + z * z;
    }

    float runmin[8];
#pragma unroll
    for (int r = 0; r < 8; ++r) runmin[r] = SENTINEL;

    for (int q = 0; q < 20; ++q) {               // x = q/20 is wave-uniform
        const float xq = (float)q * (1.0f / 20.0f);
        const float x2 = xq * xq;
        const int cBase0 = q * 400;
        for (int rb = 0; rb < 400; rb += 16) {   // 16-col tile, never straddles q or word
            const float4 e = tab[rb + nSlot];
            const int c = cBase0 + rb + nSlot;
            const unsigned w = colMask[c >> 5];  // tile-uniform LDS broadcast
            const bool inSet = (w >> (c & 31)) & 1u;
            const float bias = inSet ? 0.0f : SENTINEL;

            v2f B;
            B.x = half ? e.y : xq;                        // K2=z | K0=x
            B.y = half ? (x2 + e.z + bias) : e.x;         // K3=|c|^2+bias | K1=y

            v8f D = __builtin_amdgcn_wmma_f32_16x16x4_f32(
                false, A, false, B, (short)0, C, false, false);
#pragma unroll
            for (int r = 0; r < 8; ++r) runmin[r] = fminf(runmin[r], D[r]);
        }
    }

    // min across the 16 lanes sharing each row (xor < 16 stays within the half)
#pragma unroll
    for (int r = 0; r < 8; ++r) {
        float v = runmin[r];
        v = fminf(v, __shfl_xor(v, 1, 32));
        v = fminf(v, __shfl_xor(v, 2, 32));
        v = fminf(v, __shfl_xor(v, 4, 32));
        v = fminf(v, __shfl_xor(v, 8, 32));
        runmin[r] = v;
    }

    // lanes 0 and 16 hold full mins for rows rowBase + r + 8*half
    float localMax = 0.0f;
    if (nSlot == 0) {
        const unsigned wA = mA[rowBase >> 5];
        const unsigned wB = mB[rowBase >> 5];
        const int bitBase = rowBase & 31;     // 0 or 16
#pragma unroll
        for (int r = 0; r < 8; ++r) {
            int k = r + 8 * half;
            bool a = (wA >> (bitBase + k)) & 1u;
            bool b = (wB >> (bitBase + k)) & 1u;
            bool only = (dir == 0) ? (a && !b) : (b && !a);
            if (only) {
                float mn = sqrtf(fmaxf(runmin[r], 0.0f));
                localMax = fmaxf(localMax, mn);
            }
        }
    }
    // combine lane 0 and lane 16 (values >= 0; other lanes contribute 0)
    localMax = fmaxf(localMax, __shfl_xor(localMax, 16, 32));
    localMax = fmaxf(localMax, __shfl_xor(localMax, 8, 32));
    localMax = fmaxf(localMax, __shfl_xor(localMax, 4, 32));
    localMax = fmaxf(localMax, __shfl_xor(localMax, 2, 32));
    localMax = fmaxf(localMax, __shfl_xor(localMax, 1, 32));

    if (lane == 0) partial[sdi * PART_STRIDE + rowTile] = localMax;
}

// ---------------- kernel 3: final reduction + empty-set semantics ----------------
__global__ __launch_bounds__(256) void final_kernel(const float* __restrict__ partial,
                                                    const int* __restrict__ cnt,
                                                    float* __restrict__ out) {
    __shared__ float red[256];
    const int t = threadIdx.x;
    const int sdi = t >> 6;      // 64 threads per (sample,direction)
    const int j0 = t & 63;
    float m = 0.0f;
    for (int j = j0; j < ROWTILES; j += 64) m = fmaxf(m, partial[sdi * PART_STRIDE + j]);
    red[t] = m;
    __syncthreads();
    for (int off = 32; off >= 1; off >>= 1) {
        if ((t & 63) < off) red[t] = fmaxf(red[t], red[t + off]);
        __syncthreads();
    }
    if (t == 0) {
        float acc = 0.0f;
        for (int s = 0; s < 2; ++s) {
            float maxA = red[(s * 2 + 0) * 64];   // max over A\B of min-to-B
            float maxB = red[(s * 2 + 1) * 64];   // max over B\A of min-to-A
            int cA = cnt[s * 4 + 0];
            int cAo = cnt[s * 4 + 1];
            int cBo = cnt[s * 4 + 2];
            float dA = (cAo > 0) ? maxA : 0.0f;
            float dB = (cBo > 0) ? ((cA > 0) ? maxB : 999.0f) : 0.0f;
            acc += fmaxf(dA, dB);
        }
        out[0] = 0.5f * acc;
    }
}

extern "C" void kernel_launch(void* const* d_in, const int* in_sizes, int n_in,
                              void* d_out, int out_size, void* d_ws, size_t ws_size,
                              hipStream_t stream) {
    const float* predict = (const float*)d_in[0];
    const float* target  = (const float*)d_in[1];
    float* out = (float*)d_out;

    // workspace layout (bytes):
    //   [0,     2048)  maskA[2][256] words
    //   [2048,  4096)  maskB[2][256] words
    //   [4096,  4128)  cnt[8] ints
    //   [4128, 12320)  partial[4][512] floats
    char* ws = (char*)d_ws;
    unsigned* maskA = (unsigned*)(ws + 0);
    unsigned* maskB = (unsigned*)(ws + 2048);
    int* cnt        = (int*)(ws + 4096);
    float* partial  = (float*)(ws + 4128);

    init_kernel<<<1, 32, 0, stream>>>(cnt);

    int nThread = 2 * V;
    mask_kernel<<<(nThread + 255) / 256, 256, 0, stream>>>(predict, target, maskA, maskB, cnt);

    dim3 grid2(ROWTILES / WAVES_PB, 4);
    haus_min_kernel<<<grid2, BLOCK2, 0, stream>>>(maskA, maskB, partial);

    final_kernel<<<1, 256, 0, stream>>>(partial, cnt, out);
}